// eppConstrainer_background_28123445854331
// MI455X (gfx1250) — compile-verified
//
#include <hip/hip_runtime.h>
#include <hip/hip_bf16.h>

// ---------------------------------------------------------------------------
// eppConstrainer: loss = sum(|pts2^T E pts1| * valid) / (sum(valid) + 1)
//   flowest: [16, 2, 512, 1024] f32   E: [16, 3, 3] f32   valid: [16,512,1024]
// Memory-bound streaming reduction (12 B/pixel, ~100 MB total).
// ---------------------------------------------------------------------------

typedef float v2f __attribute__((ext_vector_type(2)));
typedef float v8f __attribute__((ext_vector_type(8)));

#define BATCH 16
#define H_    512
#define W_    1024
#define HW_   (H_ * W_)

// Full-wave (32 lane) f32 sum using V_WMMA_F32_16X16X4_F32.
// A: lane puts its value at K=0 (lanes 0-15) / K=2 (lanes 16-31), zero else.
// B: all ones  ->  D[m][n] = s_m + s_{m+16} (independent of n).
// Lane l<16 holds rows 0..7 in D VGPRs, lane l>=16 rows 8..15; sum the 8
// components then fold halves with one xor-16 shuffle.
__device__ __forceinline__ float wave_sum_f32(float s) {
    v2f a; a[0] = s;    a[1] = 0.0f;
    v2f b; b[0] = 1.0f; b[1] = 1.0f;
    v8f c = {};
    v8f d = __builtin_amdgcn_wmma_f32_16x16x4_f32(
        /*neg_a=*/false, a, /*neg_b=*/false, b,
        /*c_mod=*/(short)0, c, /*reuse_a=*/false, /*reuse_b=*/false);
    float p = ((d[0] + d[1]) + (d[2] + d[3])) + ((d[4] + d[5]) + (d[6] + d[7]));
    p += __shfl_xor(p, 16, 32);
    return p;
}

__global__ void epp_init_kernel(float* __restrict__ acc) {
    if (threadIdx.x < 2) acc[threadIdx.x] = 0.0f;
}

__global__ __launch_bounds__(256)
void epp_main_kernel(const float* __restrict__ flow,
                     const float* __restrict__ E,
                     const float* __restrict__ valid,
                     float* __restrict__ acc) {
    const int b = blockIdx.y;            // one batch per grid-Y slice
    const float* Eb = E + b * 9;         // wave-uniform -> scalar loads
    const float e00 = Eb[0], e01 = Eb[1], e02 = Eb[2];
    const float e10 = Eb[3], e11 = Eb[4], e12 = Eb[5];
    const float e20 = Eb[6], e21 = Eb[7], e22 = Eb[8];

    const float4* __restrict__ u4p = (const float4*)(flow  + (size_t)b * 2 * HW_);
    const float4* __restrict__ v4p = (const float4*)(flow  + (size_t)(b * 2 + 1) * HW_);
    const float4* __restrict__ w4p = (const float4*)(valid + (size_t)b * HW_);

    const int n4     = HW_ / 4;                       // float4 groups per batch
    const int stride = gridDim.x * blockDim.x;

    float num = 0.0f, den = 0.0f;

    for (int g = blockIdx.x * blockDim.x + threadIdx.x; g < n4; g += stride) {
        const int gn = g + stride;                    // prefetch next iteration
        if (gn < n4) {
            __builtin_prefetch(&u4p[gn], 0, 0);
            __builtin_prefetch(&v4p[gn], 0, 0);
            __builtin_prefetch(&w4p[gn], 0, 0);
        }
        const float4 u = u4p[g];
        const float4 v = v4p[g];
        const float4 w = w4p[g];

        const int   pix = g << 2;                     // 4 consecutive x, same row
        const float y   = (float)(pix >> 10);         // W_ == 1024
        const float x0  = (float)(pix & (W_ - 1));

        // y-only parts of E * pts1 (shared across the 4 pixels)
        const float ry0 = fmaf(e01, y, e02);
        const float ry1 = fmaf(e11, y, e12);
        const float ry2 = fmaf(e21, y, e22);

        {   const float x = x0;          // pixel 0
            const float r0 = fmaf(e00, x, ry0);
            const float r1 = fmaf(e10, x, ry1);
            const float r2 = fmaf(e20, x, ry2);
            const float ep = fmaf(x + u.x, r0, fmaf(y + v.x, r1, r2));
            num = fmaf(fabsf(ep), w.x, num);  den += w.x; }
        {   const float x = x0 + 1.0f;   // pixel 1
            const float r0 = fmaf(e00, x, ry0);
            const float r1 = fmaf(e10, x, ry1);
            const float r2 = fmaf(e20, x, ry2);
            const float ep = fmaf(x + u.y, r0, fmaf(y + v.y, r1, r2));
            num = fmaf(fabsf(ep), w.y, num);  den += w.y; }
        {   const float x = x0 + 2.0f;   // pixel 2
            const float r0 = fmaf(e00, x, ry0);
            const float r1 = fmaf(e10, x, ry1);
            const float r2 = fmaf(e20, x, ry2);
            const float ep = fmaf(x + u.z, r0, fmaf(y + v.z, r1, r2));
            num = fmaf(fabsf(ep), w.z, num);  den += w.z; }
        {   const float x = x0 + 3.0f;   // pixel 3
            const float r0 = fmaf(e00, x, ry0);
            const float r1 = fmaf(e10, x, ry1);
            const float r2 = fmaf(e20, x, ry2);
            const float ep = fmaf(x + u.w, r0, fmaf(y + v.w, r1, r2));
            num = fmaf(fabsf(ep), w.w, num);  den += w.w; }
    }

    // --- epilogue: EXEC is full here (reconverged after uniform-bound loop) ---
    const float wnum = wave_sum_f32(num);
    const float wden = wave_sum_f32(den);

    __shared__ float sN, sD;
    if (threadIdx.x == 0) { sN = 0.0f; sD = 0.0f; }
    __syncthreads();
    if ((threadIdx.x & 31) == 0) {       // one lane per wave -> ds_add_f32
        atomicAdd(&sN, wnum);
        atomicAdd(&sD, wden);
    }
    __syncthreads();
    if (threadIdx.x == 0) {              // one global_atomic_add_f32 per block
        atomicAdd(&acc[0], sN);
        atomicAdd(&acc[1], sD);
    }
}

__global__ void epp_finalize_kernel(const float* __restrict__ acc,
                                    float* __restrict__ out) {
    if (threadIdx.x == 0 && blockIdx.x == 0)
        out[0] = acc[0] / (acc[1] + 1.0f);
}

extern "C" void kernel_launch(void* const* d_in, const int* in_sizes, int n_in,
                              void* d_out, int out_size, void* d_ws, size_t ws_size,
                              hipStream_t stream) {
    (void)in_sizes; (void)n_in; (void)out_size; (void)ws_size;
    const float* flow  = (const float*)d_in[0];   // [16,2,512,1024]
    const float* E     = (const float*)d_in[1];   // [16,3,3]
    const float* valid = (const float*)d_in[2];   // [16,512,1024]
    float*       out   = (float*)d_out;           // scalar
    float*       acc   = (float*)d_ws;            // [num, den]

    epp_init_kernel<<<1, 32, 0, stream>>>(acc);

    dim3 grid(128, BATCH, 1);                     // 2048 blocks x 256 thr (wave32)
    dim3 block(256, 1, 1);
    epp_main_kernel<<<grid, block, 0, stream>>>(flow, E, valid, acc);

    epp_finalize_kernel<<<1, 32, 0, stream>>>(acc, out);
}